// CorrectSmoothBinaryClassifier_43490838839924
// MI455X (gfx1250) — compile-verified
//
#include <hip/hip_runtime.h>
#include <math.h>

#define N_NODES 100000
#define N_EDGES 1600000
#define DFEAT 64
#define HFEAT 64

typedef __attribute__((ext_vector_type(2))) float v2f;
typedef __attribute__((ext_vector_type(8))) float v8f;

// ---------------- degree count: cnt[dst] += 1 ----------------
__global__ __launch_bounds__(256) void init_cnt(float* cnt, int n) {
    int i = blockIdx.x * blockDim.x + threadIdx.x;
    if (i < n) cnt[i] = 0.0f;
}

__global__ __launch_bounds__(256) void count_deg(const int* __restrict__ dst, float* cnt, int e) {
    int i = blockIdx.x * blockDim.x + threadIdx.x;
    if (i < e) atomicAdd(&cnt[dst[i]], 1.0f);
}

// dinv = (cnt+1)^-0.5 ; dinvcs = cnt>0 ? cnt^-0.5 : 0
__global__ __launch_bounds__(256) void make_dinv(const float* __restrict__ cnt,
                                                 float* dinv, float* dinvcs, int n) {
    int i = blockIdx.x * blockDim.x + threadIdx.x;
    if (i < n) {
        float c = cnt[i];
        dinv[i]   = rsqrtf(c + 1.0f);
        dinvcs[i] = (c > 0.0f) ? rsqrtf(c) : 0.0f;
    }
}

// per-edge norms (computed once, reused 100+ times)
__global__ __launch_bounds__(256) void make_norms(const int* __restrict__ src, const int* __restrict__ dst,
                                                  const float* __restrict__ dinv, const float* __restrict__ dinvcs,
                                                  float* norm1, float* normcs, int e) {
    int i = blockIdx.x * blockDim.x + threadIdx.x;
    if (i < e) {
        int s = src[i], d = dst[i];
        norm1[i]  = dinv[s] * dinv[d];
        normcs[i] = dinvcs[s] * dinvcs[d];
    }
}

// ---------------- WMMA f32 GEMM: XW = X(N x 64) @ W(64 x 64) ----------------
// 256 threads = 8 waves/block; each wave computes one 16x16 tile.
// Block covers 32 rows x 64 cols (2 row-tiles x 4 col-tiles).
__global__ __launch_bounds__(256) void gemm_xw1(const float* __restrict__ X,
                                                const float* __restrict__ W,
                                                float* __restrict__ XW) {
    int wave = threadIdx.x >> 5;
    int lane = threadIdx.x & 31;
    int row0 = (blockIdx.x * 2 + (wave >> 2)) * 16;
    int col0 = (wave & 3) * 16;
    int half = lane >> 4;   // lanes 16-31 carry K+2 / M+8 per ISA layout
    int l    = lane & 15;

    v8f c = {};
#pragma unroll
    for (int k = 0; k < 64; k += 4) {
        int ka = k + 2 * half;
        v2f a, b;
        // A 16x4 f32: lane l holds row M=l; v0=K=ka, v1=K=ka+1
        a.x = X[(size_t)(row0 + l) * 64 + ka];
        a.y = X[(size_t)(row0 + l) * 64 + ka + 1];
        // B 4x16 f32: lane l holds col N=l; v0=K=ka, v1=K=ka+1
        b.x = W[(size_t)(ka)     * 64 + col0 + l];
        b.y = W[(size_t)(ka + 1) * 64 + col0 + l];
        c = __builtin_amdgcn_wmma_f32_16x16x4_f32(false, a, false, b,
                                                  (short)0, c, false, false);
    }
    // C/D layout: VGPR r, lanes 0-15: M=r ; lanes 16-31: M=r+8
#pragma unroll
    for (int r = 0; r < 8; ++r) {
        int row = row0 + r + 8 * half;
        XW[(size_t)row * 64 + col0 + l] = c[r];
    }
}

// ---------------- 64-channel edge scatter: agg64[dst,:] += XW[src,:]*norm ----------------
__global__ __launch_bounds__(256) void zero_f(float* p, int n) {
    int i = blockIdx.x * blockDim.x + threadIdx.x;
    if (i < n) p[i] = 0.0f;
}

__global__ __launch_bounds__(256) void conv64_scatter(const int* __restrict__ src, const int* __restrict__ dst,
                                                      const float* __restrict__ norm1,
                                                      const float* __restrict__ XW,
                                                      float* agg64, int e) {
    int t = blockIdx.x * blockDim.x + threadIdx.x;
    int ed = t >> 6;
    int ch = t & 63;
    if (ed < e) {
        int s = src[ed], d = dst[ed];
        atomicAdd(&agg64[(size_t)d * 64 + ch], XW[(size_t)s * 64 + ch] * norm1[ed]);
    }
}

// h = relu(agg64 + XW*dinv^2 + b1)  (h overwrites agg64)
__global__ __launch_bounds__(256) void relu_self(const float* __restrict__ XW,
                                                 const float* __restrict__ dinv,
                                                 const float* __restrict__ b1,
                                                 float* agg64, int n64) {
    int t = blockIdx.x * blockDim.x + threadIdx.x;
    if (t < n64) {
        int i = t >> 6, ch = t & 63;
        float di = dinv[i];
        float v = agg64[t] + XW[t] * di * di + b1[ch];
        agg64[t] = v > 0.0f ? v : 0.0f;
    }
}

// hw2[i] = dot(h[i,:64], W2)  — one wave32 per row, two channels per lane
__global__ __launch_bounds__(256) void dot_w2(const float* __restrict__ h,
                                              const float* __restrict__ W2,
                                              float* hw2, int n) {
    int gw = (blockIdx.x * blockDim.x + threadIdx.x) >> 5;
    int lane = threadIdx.x & 31;
    if (gw < n) {
        float s = h[(size_t)gw * 64 + lane] * W2[lane]
                + h[(size_t)gw * 64 + 32 + lane] * W2[32 + lane];
        for (int off = 16; off > 0; off >>= 1) s += __shfl_down(s, off);
        if (lane == 0) hw2[gw] = s;
    }
}

// 1-channel edge scatter
__global__ __launch_bounds__(256) void conv1_scatter(const int* __restrict__ src, const int* __restrict__ dst,
                                                     const float* __restrict__ norm1,
                                                     const float* __restrict__ val,
                                                     float* agg, int e) {
    int i = blockIdx.x * blockDim.x + threadIdx.x;
    if (i < e) atomicAdd(&agg[dst[i]], val[src[i]] * norm1[i]);
}

// sigmoid -> probs, error, init corr-prop state (out=error, res=(1-0.5)*error), zero agg2
__global__ __launch_bounds__(256) void probs_error(const float* __restrict__ aggL,
                                                   const float* __restrict__ hw2,
                                                   const float* __restrict__ dinv,
                                                   const unsigned char* __restrict__ mask,
                                                   const int* __restrict__ labels,
                                                   const float* __restrict__ b2,
                                                   float* probs, float* error, float* res,
                                                   float* out, float* agg2, int n) {
    int i = blockIdx.x * blockDim.x + threadIdx.x;
    if (i < n) {
        float di = dinv[i];
        float logit = aggL[i] + hw2[i] * di * di + b2[0];
        float p = 1.0f / (1.0f + __expf(-logit));
        float p0 = 1.0f - p;
        probs[2 * i] = p0; probs[2 * i + 1] = p;
        bool m = mask[i] != 0;
        int lab = labels[i];
        float o0 = (lab == 0) ? 1.0f : 0.0f;
        float o1 = (lab == 1) ? 1.0f : 0.0f;
        float e0 = m ? (o0 - p0) : 0.0f;
        float e1 = m ? (o1 - p)  : 0.0f;
        error[2 * i] = e0; error[2 * i + 1] = e1;
        res[2 * i] = 0.5f * e0; res[2 * i + 1] = 0.5f * e1;   // (1-A_CORR)*error
        out[2 * i] = e0; out[2 * i + 1] = e1;                 // x0 = error
        agg2[2 * i] = 0.0f; agg2[2 * i + 1] = 0.0f;
    }
}

// 2-channel edge scatter for label prop
__global__ __launch_bounds__(256) void scatter2(const int* __restrict__ src, const int* __restrict__ dst,
                                                const float* __restrict__ normcs,
                                                const float* __restrict__ out,
                                                float* agg2, int e) {
    int i = blockIdx.x * blockDim.x + threadIdx.x;
    if (i < e) {
        int s = src[i], d = dst[i];
        float nm = normcs[i];
        float2 v = ((const float2*)out)[s];
        atomicAdd(&agg2[2 * d],     v.x * nm);
        atomicAdd(&agg2[2 * d + 1], v.y * nm);
    }
}

// correct step: out = mask ? error : (0.5*agg2 + res); agg2 = 0 (fused re-zero)
__global__ __launch_bounds__(256) void corr_step(const unsigned char* __restrict__ mask,
                                                 const float* __restrict__ error,
                                                 const float* __restrict__ res,
                                                 float* out, float* agg2, int n) {
    int i = blockIdx.x * blockDim.x + threadIdx.x;
    if (i < n) {
        bool m = mask[i] != 0;
        float v0 = 0.5f * agg2[2 * i]     + res[2 * i];
        float v1 = 0.5f * agg2[2 * i + 1] + res[2 * i + 1];
        out[2 * i]     = m ? error[2 * i]     : v0;
        out[2 * i + 1] = m ? error[2 * i + 1] : v1;
        agg2[2 * i] = 0.0f; agg2[2 * i + 1] = 0.0f;
    }
}

// corrected = probs + out ; y = mask ? onehot : corrected ; out=y ; res=(1-0.8)*y
__global__ __launch_bounds__(256) void make_y(const unsigned char* __restrict__ mask,
                                              const int* __restrict__ labels,
                                              const float* __restrict__ probs,
                                              float* out, float* res, int n) {
    int i = blockIdx.x * blockDim.x + threadIdx.x;
    if (i < n) {
        bool m = mask[i] != 0;
        int lab = labels[i];
        float o0 = (lab == 0) ? 1.0f : 0.0f;
        float o1 = (lab == 1) ? 1.0f : 0.0f;
        float c0 = probs[2 * i]     + out[2 * i];
        float c1 = probs[2 * i + 1] + out[2 * i + 1];
        float y0 = m ? o0 : c0;
        float y1 = m ? o1 : c1;
        out[2 * i] = y0; out[2 * i + 1] = y1;
        res[2 * i] = 0.2f * y0; res[2 * i + 1] = 0.2f * y1;
    }
}

// smooth step: out = clamp(0.8*agg2 + res, 0, 1); agg2 = 0
__global__ __launch_bounds__(256) void smooth_step(const float* __restrict__ res,
                                                   float* out, float* agg2, int n) {
    int i = blockIdx.x * blockDim.x + threadIdx.x;
    if (i < n) {
        float v0 = 0.8f * agg2[2 * i]     + res[2 * i];
        float v1 = 0.8f * agg2[2 * i + 1] + res[2 * i + 1];
        v0 = fminf(fmaxf(v0, 0.0f), 1.0f);
        v1 = fminf(fmaxf(v1, 0.0f), 1.0f);
        out[2 * i] = v0; out[2 * i + 1] = v1;
        agg2[2 * i] = 0.0f; agg2[2 * i + 1] = 0.0f;
    }
}

__global__ __launch_bounds__(256) void final_logit(const float* __restrict__ out,
                                                   float* __restrict__ y, int n) {
    int i = blockIdx.x * blockDim.x + threadIdx.x;
    if (i < n) y[i] = logf(out[2 * i + 1] + 1e-12f) - logf(out[2 * i] + 1e-12f);
}

// -----------------------------------------------------------------------------
extern "C" void kernel_launch(void* const* d_in, const int* in_sizes, int n_in,
                              void* d_out, int out_size, void* d_ws, size_t ws_size,
                              hipStream_t stream) {
    const int N = N_NODES, E = N_EDGES;

    const float*         x      = (const float*)d_in[0];
    const int*           eidx   = (const int*)d_in[1];          // [2, E]
    const unsigned char* mask   = (const unsigned char*)d_in[2];
    const int*           labels = (const int*)d_in[3];
    const float*         W1     = (const float*)d_in[4];
    const float*         b1     = (const float*)d_in[5];
    const float*         W2     = (const float*)d_in[6];
    const float*         b2     = (const float*)d_in[7];

    const int* src = eidx;
    const int* dst = eidx + E;

    // --- carve workspace ---
    char* ws = (char*)d_ws;
    size_t off = 0;
    auto carve = [&](size_t bytes) {
        void* p = ws + off;
        off = (off + bytes + 255) & ~(size_t)255;
        return p;
    };
    float* cnt    = (float*)carve((size_t)N * 4);
    float* dinv   = (float*)carve((size_t)N * 4);
    float* dinvcs = (float*)carve((size_t)N * 4);
    float* norm1  = (float*)carve((size_t)E * 4);
    float* normcs = (float*)carve((size_t)E * 4);
    float* XW     = (float*)carve((size_t)N * 64 * 4);
    float* agg64  = (float*)carve((size_t)N * 64 * 4);   // later holds h
    float* hw2    = (float*)carve((size_t)N * 4);
    float* aggL   = (float*)carve((size_t)N * 4);
    float* probs  = (float*)carve((size_t)N * 2 * 4);
    float* error  = (float*)carve((size_t)N * 2 * 4);
    float* res    = (float*)carve((size_t)N * 2 * 4);
    float* out    = (float*)carve((size_t)N * 2 * 4);
    float* agg2   = (float*)carve((size_t)N * 2 * 4);

    const int T = 256;
    dim3 blk(T);
    unsigned gN   = (N + T - 1) / T;
    unsigned gE   = (E + T - 1) / T;
    unsigned gN64 = ((unsigned)((size_t)N * 64 + T - 1)) / T;
    unsigned gE64 = (unsigned)(((size_t)E * 64 + T - 1) / T);

    // degrees + norms
    init_cnt<<<gN, blk, 0, stream>>>(cnt, N);
    count_deg<<<gE, blk, 0, stream>>>(dst, cnt, E);
    make_dinv<<<gN, blk, 0, stream>>>(cnt, dinv, dinvcs, N);
    make_norms<<<gE, blk, 0, stream>>>(src, dst, dinv, dinvcs, norm1, normcs, E);

    // XW = x @ W1 via V_WMMA_F32_16X16X4_F32 (N=100000 divisible by 32)
    gemm_xw1<<<N / 32, blk, 0, stream>>>(x, W1, XW);

    // GCN conv 1 + relu
    zero_f<<<gN64, blk, 0, stream>>>(agg64, N * 64);
    conv64_scatter<<<gE64, blk, 0, stream>>>(src, dst, norm1, XW, agg64, E);
    relu_self<<<gN64, blk, 0, stream>>>(XW, dinv, b1, agg64, N * 64);   // agg64 := h

    // h @ W2 (wave per row), then GCN conv 2 (1 channel)
    dot_w2<<<(unsigned)(((size_t)N * 32 + T - 1) / T), blk, 0, stream>>>(agg64, W2, hw2, N);
    zero_f<<<gN, blk, 0, stream>>>(aggL, N);
    conv1_scatter<<<gE, blk, 0, stream>>>(src, dst, norm1, hw2, aggL, E);

    // probs, error, init correction-prop state
    probs_error<<<gN, blk, 0, stream>>>(aggL, hw2, dinv, mask, labels, b2,
                                        probs, error, res, out, agg2, N);

    // correction propagation: 50 iters, alpha=0.5, post=fix
    for (int it = 0; it < 50; ++it) {
        scatter2<<<gE, blk, 0, stream>>>(src, dst, normcs, out, agg2, E);
        corr_step<<<gN, blk, 0, stream>>>(mask, error, res, out, agg2, N);
    }

    // y = mask ? onehot : probs + smoothed_error ; init smooth-prop state
    make_y<<<gN, blk, 0, stream>>>(mask, labels, probs, out, res, N);

    // smoothing propagation: 50 iters, alpha=0.8, post=clamp
    for (int it = 0; it < 50; ++it) {
        scatter2<<<gE, blk, 0, stream>>>(src, dst, normcs, out, agg2, E);
        smooth_step<<<gN, blk, 0, stream>>>(res, out, agg2, N);
    }

    final_logit<<<gN, blk, 0, stream>>>(out, (float*)d_out, N);
}